// GNNGM1_94489280746
// MI455X (gfx1250) — compile-verified
//
#include <hip/hip_runtime.h>
#include <hip/hip_bf16.h>
#include <math.h>

typedef __attribute__((ext_vector_type(2))) float v2f;
typedef __attribute__((ext_vector_type(8))) float v8f;

#define N1 1024
#define NEL (N1 * N1)

__device__ __forceinline__ float nan_fix(float v) {
    if (v != v) return 0.0f;
    if (v == __builtin_inff()) return 3.4028235e38f;
    if (v == -__builtin_inff()) return -3.4028235e38f;
    return v;
}

// ---------- degrees: row sums of a 1024x1024 matrix ----------
__global__ __launch_bounds__(256) void degree_k(const float* __restrict__ G, float* __restrict__ d) {
    __shared__ float red[256];
    const int row = blockIdx.x, tid = threadIdx.x;
    float s = 0.0f;
    for (int j = tid; j < N1; j += 256) s += G[row * N1 + j];
    red[tid] = s; __syncthreads();
    for (int st = 128; st > 0; st >>= 1) { if (tid < st) red[tid] += red[tid + st]; __syncthreads(); }
    if (tid == 0) d[row] = red[0];
}

// ---------- ttheta = mean(d1) ----------
__global__ __launch_bounds__(256) void mean_k(const float* __restrict__ d, float* __restrict__ tt) {
    __shared__ float red[256];
    const int tid = threadIdx.x;
    red[tid] = d[tid] + d[tid + 256] + d[tid + 512] + d[tid + 768];
    __syncthreads();
    for (int st = 128; st > 0; st >>= 1) { if (tid < st) red[tid] += red[tid + st]; __syncthreads(); }
    if (tid == 0) tt[0] = red[0] / 1024.0f;
}

// ---------- S0[i,j] = -|d1[i]-d2[j]| ----------
__global__ __launch_bounds__(256) void s0_k(const float* __restrict__ d1, const float* __restrict__ d2,
                                            float* __restrict__ S) {
    const int idx = blockIdx.x * 256 + threadIdx.x;
    const int i = idx >> 10, j = idx & (N1 - 1);
    S[idx] = -fabsf(d1[i] - d2[j]);
}

// ---------- elementwise MLP: relu(s*W1+b1)@W2 + b2 ----------
__global__ __launch_bounds__(256) void mlp_k(const float* __restrict__ S, float* __restrict__ Mo,
                                             const float* __restrict__ W1, const float* __restrict__ b1,
                                             const float* __restrict__ W2, const float* __restrict__ b2,
                                             const float* __restrict__ tt, int useTheta) {
    const int idx = blockIdx.x * 256 + threadIdx.x;
    float s = S[idx];
    if (useTheta) s /= tt[0];
    float acc = b2[0];
#pragma unroll 8
    for (int h = 0; h < 64; ++h) {
        float hv = fmaxf(s * W1[h] + b1[h], 0.0f);
        acc += hv * W2[h];
    }
    Mo[idx] = acc;
}

// ---------- row softmax (axis=1), optional nan_to_num ----------
__global__ __launch_bounds__(256) void row_softmax_k(const float* __restrict__ X, float* __restrict__ Y,
                                                     int guard) {
    __shared__ float red[256];
    const int row = blockIdx.x, tid = threadIdx.x;
    const float* x = X + row * N1;
    float m = -3.4028235e38f;
    for (int j = tid; j < N1; j += 256) m = fmaxf(m, x[j]);
    red[tid] = m; __syncthreads();
    for (int st = 128; st > 0; st >>= 1) { if (tid < st) red[tid] = fmaxf(red[tid], red[tid + st]); __syncthreads(); }
    const float rmax = red[0]; __syncthreads();
    float s = 0.0f;
    for (int j = tid; j < N1; j += 256) s += expf(x[j] - rmax);
    red[tid] = s; __syncthreads();
    for (int st = 128; st > 0; st >>= 1) { if (tid < st) red[tid] += red[tid + st]; __syncthreads(); }
    const float rs = red[0];
    for (int j = tid; j < N1; j += 256) {
        float v = expf(x[j] - rmax) / rs;
        if (guard) v = nan_fix(v);
        Y[row * N1 + j] = v;
    }
}

// ---------- row normalize in place (divide by row sum) ----------
__global__ __launch_bounds__(256) void rownorm_k(float* __restrict__ P) {
    __shared__ float red[256];
    const int row = blockIdx.x, tid = threadIdx.x;
    float* x = P + row * N1;
    float s = 0.0f;
    for (int j = tid; j < N1; j += 256) s += x[j];
    red[tid] = s; __syncthreads();
    for (int st = 128; st > 0; st >>= 1) { if (tid < st) red[tid] += red[tid + st]; __syncthreads(); }
    const float rs = red[0];
    for (int j = tid; j < N1; j += 256) x[j] = x[j] / rs;
}

// ---------- column sums ----------
__global__ __launch_bounds__(256) void colsum_k(const float* __restrict__ P, float* __restrict__ cs) {
    const int j = blockIdx.x * 256 + threadIdx.x;
    float s = 0.0f;
    for (int i = 0; i < N1; ++i) s += P[i * N1 + j];
    cs[j] = s;
}

__global__ __launch_bounds__(256) void coldiv_k(float* __restrict__ P, const float* __restrict__ cs) {
    const int idx = blockIdx.x * 256 + threadIdx.x;
    P[idx] = P[idx] / cs[idx & (N1 - 1)];
}

// ---------- W = log(nan_to_num(P)*1536), clamp +-inf ----------
__global__ __launch_bounds__(256) void log_k(float* __restrict__ P) {
    const int idx = blockIdx.x * 256 + threadIdx.x;
    float v = nan_fix(P[idx]);
    float w = logf(v * 1536.0f);
    if (w == -__builtin_inff()) w = -1.0e12f;
    else if (w == __builtin_inff()) w = 1.0e12f;
    P[idx] = w;
}

// ---------- build one-hot permutation B operand in WMMA lane layout ----------
// B[(nt*7+kt)*32+lane][e] = P1H[p=nt*16+(lane&15)][r][c] for k=kt*4+(lane>=16?2:0)+e, k=r*5+c
__global__ __launch_bounds__(256) void binit_k(float* __restrict__ B) {
    const int idx = blockIdx.x * 256 + threadIdx.x;
    if (idx >= 8 * 7 * 32) return;
    const int lane = idx & 31;
    const int kt = (idx >> 5) % 7;
    const int nt = idx / (7 * 32);
    const int p = nt * 16 + (lane & 15);
    const int kbase = kt * 4 + ((lane >= 16) ? 2 : 0);
    int perm[5] = {0, 0, 0, 0, 0};
    if (p < 120) {  // Lehmer decode, lexicographic (matches itertools.permutations)
        int avail[5] = {0, 1, 2, 3, 4};
        const int fact[5] = {24, 6, 2, 1, 1};
        int rem = p;
        for (int i = 0; i < 5; ++i) {
            int dd = rem / fact[i]; rem -= dd * fact[i];
            perm[i] = avail[dd];
            for (int q = dd; q < 4 - i; ++q) avail[q] = avail[q + 1];
        }
    }
    for (int e = 0; e < 2; ++e) {
        const int k = kbase + e;
        float v = 0.0f;
        if (p < 120 && k < 25) {
            const int r = k / 5, c = k - 5 * r;
            v = (perm[r] == c) ? 1.0f : 0.0f;
        }
        B[idx * 2 + e] = v;
    }
}

// ---------- assignment: X1[a,b] = max over 120 perms of sum_r W[adj1[a,r], adj2[b,perm(r)]] ----------
// scores = small[16 pairs, 25pad28] x P1H^T[28, 120pad128] via V_WMMA_F32_16X16X4_F32
__global__ __launch_bounds__(256) void assign_k(const float* __restrict__ W,
                                                const int* __restrict__ adj1,
                                                const int* __restrict__ adj2,
                                                const float* __restrict__ Bg,
                                                float* __restrict__ X1) {
    __shared__ float ldsW[5 * N1];      // 5 gathered rows of W
    __shared__ float ldsB[8 * 7 * 32 * 2];
    const int tid = threadIdx.x;
    const int a = blockIdx.y;
#pragma unroll
    for (int t = 0; t < 20; ++t) {
        const int idx = tid + t * 256;
        const int r = idx >> 10, col = idx & (N1 - 1);
        ldsW[idx] = W[adj1[a * 5 + r] * N1 + col];
    }
#pragma unroll
    for (int t = 0; t < 14; ++t) {
        const int idx = tid + t * 256;
        ldsB[idx] = Bg[idx];
    }
    __syncthreads();

    const int lane = tid & 31;
    const int wave = tid >> 5;
    const int nloc = lane & 15;
    const bool hi = lane >= 16;
    const int bbase = blockIdx.x * 128 + wave * 16;
    const int b = bbase + nloc;
    const int j0 = adj2[b * 5 + 0], j1 = adj2[b * 5 + 1], j2 = adj2[b * 5 + 2],
              j3 = adj2[b * 5 + 3], j4 = adj2[b * 5 + 4];

    auto jc = [&](int c) { return c == 0 ? j0 : c == 1 ? j1 : c == 2 ? j2 : c == 3 ? j3 : j4; };
    auto ldA = [&](int k) -> float {
        if (k >= 25) return 0.0f;
        const int r = k / 5, c = k - 5 * r;
        return ldsW[r * N1 + jc(c)];
    };

    // A operand per k-tile: lanes 0-15 hold K={4kt,4kt+1}, lanes 16-31 hold K={4kt+2,4kt+3}
    v2f Aop[7];
#pragma unroll
    for (int kt = 0; kt < 7; ++kt) {
        const float xlo = ldA(4 * kt + 0), ylo = ldA(4 * kt + 1);
        const float xhi = ldA(4 * kt + 2), yhi = ldA(4 * kt + 3);
        Aop[kt].x = hi ? xhi : xlo;
        Aop[kt].y = hi ? yhi : ylo;
    }

    float vmax[8];
#pragma unroll
    for (int v = 0; v < 8; ++v) vmax[v] = -3.0e38f;

#pragma unroll
    for (int nt = 0; nt < 8; ++nt) {
        // padded columns (perm index >= 120, in tile 7) start at -1e30 so they never win
        const float cinit = (nt == 7 && nloc >= 8) ? -1.0e30f : 0.0f;
        v8f acc;
#pragma unroll
        for (int v = 0; v < 8; ++v) acc[v] = cinit;
#pragma unroll
        for (int kt = 0; kt < 7; ++kt) {
            const int bi = ((nt * 7 + kt) * 32 + lane) * 2;
            v2f Bop;
            Bop.x = ldsB[bi];
            Bop.y = ldsB[bi + 1];
            acc = __builtin_amdgcn_wmma_f32_16x16x4_f32(false, Aop[kt], false, Bop,
                                                        (short)0, acc, false, false);
        }
#pragma unroll
        for (int v = 0; v < 8; ++v) vmax[v] = fmaxf(vmax[v], acc[v]);
    }
    // D layout: VGPR v -> (M=v, N=lane) for lanes 0-15, (M=v+8, N=lane-16) for lanes 16-31.
    // Reduce max over N within each 16-lane half.
#pragma unroll
    for (int off = 8; off >= 1; off >>= 1) {
#pragma unroll
        for (int v = 0; v < 8; ++v) vmax[v] = fmaxf(vmax[v], __shfl_xor(vmax[v], off, 32));
    }
    if (lane == 0) {
#pragma unroll
        for (int v = 0; v < 8; ++v) X1[a * N1 + bbase + v] = vmax[v];
    } else if (lane == 16) {
#pragma unroll
        for (int v = 0; v < 8; ++v) X1[a * N1 + bbase + 8 + v] = vmax[v];
    }
}

// ---------- column softmax stats for final masked softmax ----------
__global__ __launch_bounds__(256) void colstats_k(const float* __restrict__ T,
                                                  float* __restrict__ cmax, float* __restrict__ csum) {
    const int j = blockIdx.x * 256 + threadIdx.x;
    float m = -3.4028235e38f;
    for (int i = 0; i < N1; ++i) m = fmaxf(m, T[i * N1 + j]);
    float s = 0.0f;
    for (int i = 0; i < N1; ++i) s += expf(T[i * N1 + j] - m);
    cmax[j] = m;
    csum[j] = s;
}

// ---------- out = 0.5*(rowSoftmax + colSoftmax), both nan_to_num'ed ----------
__global__ __launch_bounds__(256) void combine_k(const float* __restrict__ A, const float* __restrict__ T,
                                                 const float* __restrict__ cmax, const float* __restrict__ csum,
                                                 float* __restrict__ out) {
    const int idx = blockIdx.x * 256 + threadIdx.x;
    const int j = idx & (N1 - 1);
    float v = expf(T[idx] - cmax[j]) / csum[j];
    v = nan_fix(v);
    out[idx] = 0.5f * (A[idx] + v);
}

extern "C" void kernel_launch(void* const* d_in, const int* in_sizes, int n_in,
                              void* d_out, int out_size, void* d_ws, size_t ws_size,
                              hipStream_t stream) {
    const float* G1   = (const float*)d_in[0];
    const float* G2   = (const float*)d_in[1];
    const int*   adj1 = (const int*)d_in[2];
    const int*   adj2 = (const int*)d_in[3];
    const float* roW1 = (const float*)d_in[4];   // [3,1,64]
    const float* rob1 = (const float*)d_in[5];   // [3,64]
    const float* roW2 = (const float*)d_in[6];   // [3,64,1]
    const float* rob2 = (const float*)d_in[7];   // [3,1]
    const float* fW1  = (const float*)d_in[8];
    const float* fb1  = (const float*)d_in[9];
    const float* fW2  = (const float*)d_in[10];
    const float* fb2  = (const float*)d_in[11];
    float* out = (float*)d_out;

    char* ws = (char*)d_ws;
    float* S = (float*)(ws);                    // 4 MB
    float* M = (float*)(ws + (4u << 20));       // 4 MB
    float* P = (float*)(ws + (8u << 20));       // 4 MB (sinkhorn -> W in place)
    float* A = (float*)(ws + (12u << 20));      // 4 MB
    char* V  = ws + (16u << 20);
    float* d1   = (float*)(V);
    float* d2   = (float*)(V + 4096);
    float* tt   = (float*)(V + 8192);
    float* cs   = (float*)(V + 12288);
    float* cmax = (float*)(V + 16384);
    float* csum = (float*)(V + 20480);
    float* Bm   = (float*)(V + 24576);          // 14336 B

    degree_k<<<N1, 256, 0, stream>>>(G1, d1);
    degree_k<<<N1, 256, 0, stream>>>(G2, d2);
    mean_k<<<1, 256, 0, stream>>>(d1, tt);
    binit_k<<<7, 256, 0, stream>>>(Bm);
    s0_k<<<NEL / 256, 256, 0, stream>>>(d1, d2, S);

    for (int li = 0; li < 3; ++li) {
        mlp_k<<<NEL / 256, 256, 0, stream>>>(S, M, roW1 + 64 * li, rob1 + 64 * li,
                                             roW2 + 64 * li, rob2 + li, tt, 1);
        row_softmax_k<<<N1, 256, 0, stream>>>(M, P, 0);
        // sinkhorn iterations: axis 0,1,0,1,0
        colsum_k<<<4, 256, 0, stream>>>(P, cs);
        coldiv_k<<<NEL / 256, 256, 0, stream>>>(P, cs);
        rownorm_k<<<N1, 256, 0, stream>>>(P);
        colsum_k<<<4, 256, 0, stream>>>(P, cs);
        coldiv_k<<<NEL / 256, 256, 0, stream>>>(P, cs);
        rownorm_k<<<N1, 256, 0, stream>>>(P);
        colsum_k<<<4, 256, 0, stream>>>(P, cs);
        coldiv_k<<<NEL / 256, 256, 0, stream>>>(P, cs);
        log_k<<<NEL / 256, 256, 0, stream>>>(P);
        assign_k<<<dim3(8, N1), 256, 0, stream>>>(P, adj1, adj2, Bm, S);
    }

    mlp_k<<<NEL / 256, 256, 0, stream>>>(S, M, fW1, fb1, fW2, fb2, tt, 0);
    row_softmax_k<<<N1, 256, 0, stream>>>(M, A, 1);
    colstats_k<<<4, 256, 0, stream>>>(M, cmax, csum);
    combine_k<<<NEL / 256, 256, 0, stream>>>(A, M, cmax, csum, out);
}